// EGNNLayer_86208583565931
// MI455X (gfx1250) — compile-verified
//
#include <hip/hip_runtime.h>
#include <hip/hip_bf16.h>

typedef __attribute__((ext_vector_type(2))) float v2f;
typedef __attribute__((ext_vector_type(8))) float v8f;

#define DIM 64

// ---------------------------------------------------------------------------
// Kernel 1: out[0..N*64) = h, out[N*64..N*67) = x  (accumulation targets)
// ---------------------------------------------------------------------------
__global__ __launch_bounds__(256) void egnn_init_out(
    const float* __restrict__ h, const float* __restrict__ x,
    float* __restrict__ out, int n_nodes) {
  const long long total_h = (long long)n_nodes * DIM;
  const long long total   = total_h + (long long)n_nodes * 3;
  long long i = (long long)blockIdx.x * blockDim.x + threadIdx.x;
  if (i < total_h)      out[i] = h[i];
  else if (i < total)   out[i] = x[i - total_h];
}

// ---------------------------------------------------------------------------
// Kernel 2: M = relu(h @ W_h + b_h)  via V_WMMA_F32_16X16X4_F32.
// Block = 128 threads = 4 waves; block owns a 16-node M-tile, each wave owns
// one 16-wide N-tile. K=64 swept in 16 steps of 4.
//
// f32 WMMA fragment layouts (ISA 7.12.2):
//   A 16x4 : lane l<16 -> row l, K={k0,k0+1}; lane>=16 -> row l-16, K={k0+2,k0+3}
//   B 4x16 : same K split by lane-half, N = lane%16
//   C/D    : vgpr i -> row (i + 8*(lane>=16)), col lane%16
// ---------------------------------------------------------------------------
__global__ __launch_bounds__(128) void egnn_node_gemm(
    const float* __restrict__ h, const float* __restrict__ Wh,
    const float* __restrict__ bh, float* __restrict__ Mout, int n_nodes) {
  const int lane  = threadIdx.x & 31;
  const int wave  = threadIdx.x >> 5;   // N-tile 0..3
  const int l16   = lane & 15;
  const int half  = lane >> 4;          // 0 or 1
  const int mbase = blockIdx.x * 16;
  const int nbase = wave * 16;

  // Clamp the A row so EXEC stays all-ones at the WMMA (ISA requirement).
  const int mrow = min(mbase + l16, n_nodes - 1);
  const float* __restrict__ arow = h + (size_t)mrow * DIM;

  v8f c = {};
#pragma unroll
  for (int k0 = 0; k0 < DIM; k0 += 4) {
    const int ka = k0 + 2 * half;
    v2f a;
    a.x = arow[ka];
    a.y = arow[ka + 1];
    v2f b;
    b.x = Wh[(size_t)ka * DIM + nbase + l16];
    b.y = Wh[(size_t)(ka + 1) * DIM + nbase + l16];
    // D = A*B + C   (8 args: neg_a, A, neg_b, B, c_mod, C, reuse_a, reuse_b)
    c = __builtin_amdgcn_wmma_f32_16x16x4_f32(false, a, false, b,
                                              (short)0, c, false, false);
  }

  const int   ncol = nbase + l16;
  const float bias = bh[ncol];
#pragma unroll
  for (int i = 0; i < 8; ++i) {
    const int m = mbase + i + 8 * half;
    if (m < n_nodes) {
      const float v = c[i] + bias;
      Mout[(size_t)m * DIM + ncol] = fmaxf(v, 0.0f);
    }
  }
}

// ---------------------------------------------------------------------------
// Kernel 3: wnode = relu(h @ W_x + b_x)   (GEMV, tiny: 6.4 MFLOP)
// ---------------------------------------------------------------------------
__global__ __launch_bounds__(256) void egnn_node_gate(
    const float* __restrict__ h, const float* __restrict__ Wx,
    const float* __restrict__ bx, float* __restrict__ wnode, int n_nodes) {
  const int i = blockIdx.x * blockDim.x + threadIdx.x;
  if (i >= n_nodes) return;
  float acc = bx[0];
  const float* __restrict__ row = h + (size_t)i * DIM;
#pragma unroll 8
  for (int k = 0; k < DIM; ++k) acc = fmaf(row[k], Wx[k], acc);
  wnode[i] = fmaxf(acc, 0.0f);
}

// ---------------------------------------------------------------------------
// Kernel 4: out_h[row[e]] += M[col[e]]   (16 lanes per edge, float4 per lane)
// All traffic is L2-resident (M 12.8 MB, out_h 12.8 MB << 192 MB L2).
// ---------------------------------------------------------------------------
__global__ __launch_bounds__(256) void egnn_edge_h(
    const long long* __restrict__ row, const long long* __restrict__ col,
    const float* __restrict__ M, float* __restrict__ out_h, int n_edges) {
  const long long gid = (long long)blockIdx.x * blockDim.x + threadIdx.x;
  const long long e   = gid >> 4;
  const int       q   = (int)(gid & 15);
  if (e >= n_edges) return;
  const int r = (int)row[e];
  const int c = (int)col[e];
  const float4 v = *(const float4*)(M + (size_t)c * DIM + q * 4);
  float* dst = out_h + (size_t)r * DIM + q * 4;
  atomicAdd(dst + 0, v.x);
  atomicAdd(dst + 1, v.y);
  atomicAdd(dst + 2, v.z);
  atomicAdd(dst + 3, v.w);
}

// ---------------------------------------------------------------------------
// Kernel 5: out_x[row[e]] += wnode[col[e]] * (x[row[e]] - x[col[e]])
// ---------------------------------------------------------------------------
__global__ __launch_bounds__(256) void egnn_edge_x(
    const long long* __restrict__ row, const long long* __restrict__ col,
    const float* __restrict__ x, const float* __restrict__ wnode,
    float* __restrict__ out_x, int n_edges) {
  const long long e = (long long)blockIdx.x * blockDim.x + threadIdx.x;
  if (e >= n_edges) return;
  const int r = (int)row[e];
  const int c = (int)col[e];
  const float w = wnode[c];
  const float* xr = x + (size_t)r * 3;
  const float* xc = x + (size_t)c * 3;
  float* dst = out_x + (size_t)r * 3;
  atomicAdd(dst + 0, w * (xr[0] - xc[0]));
  atomicAdd(dst + 1, w * (xr[1] - xc[1]));
  atomicAdd(dst + 2, w * (xr[2] - xc[2]));
}

// ---------------------------------------------------------------------------
extern "C" void kernel_launch(void* const* d_in, const int* in_sizes, int n_in,
                              void* d_out, int out_size, void* d_ws, size_t ws_size,
                              hipStream_t stream) {
  const float*     h  = (const float*)d_in[0];
  const float*     x  = (const float*)d_in[1];
  const long long* ei = (const long long*)d_in[2];
  const float*     Wh = (const float*)d_in[3];
  const float*     bh = (const float*)d_in[4];
  const float*     Wx = (const float*)d_in[5];
  const float*     bx = (const float*)d_in[6];

  const int n_nodes = in_sizes[0] / DIM;
  const int n_edges = in_sizes[2] / 2;
  const long long* row = ei;
  const long long* col = ei + n_edges;

  float* out   = (float*)d_out;
  float* out_h = out;                               // [N, 64]
  float* out_x = out + (size_t)n_nodes * DIM;       // [N, 3]

  float* Mbuf  = (float*)d_ws;                      // [N, 64]
  float* wnode = Mbuf + (size_t)n_nodes * DIM;      // [N]

  // 1) seed outputs with h / x (scatter accumulates in place)
  {
    const long long total = (long long)n_nodes * (DIM + 3);
    const int grid = (int)((total + 255) / 256);
    egnn_init_out<<<grid, 256, 0, stream>>>(h, x, out, n_nodes);
  }
  // 2) node-level GEMM via WMMA (h[col]@W commutes with the gather)
  egnn_node_gemm<<<(n_nodes + 15) / 16, 128, 0, stream>>>(h, Wh, bh, Mbuf, n_nodes);
  // 3) node-level gate GEMV
  egnn_node_gate<<<(n_nodes + 255) / 256, 256, 0, stream>>>(h, Wx, bx, wnode, n_nodes);
  // 4) edge scatter of messages (16 lanes/edge, float4 gather + f32 atomics in L2)
  {
    const long long work = (long long)n_edges * 16;
    const int grid = (int)((work + 255) / 256);
    egnn_edge_h<<<grid, 256, 0, stream>>>(row, col, Mbuf, out_h, n_edges);
  }
  // 5) edge scatter of gated coordinate deltas
  egnn_edge_x<<<(n_edges + 255) / 256, 256, 0, stream>>>(row, col, x, wnode, out_x, n_edges);
}